// Parallel_45664092291588
// MI455X (gfx1250) — compile-verified
//
#include <hip/hip_runtime.h>
#include <math.h>

typedef __attribute__((ext_vector_type(2))) float v2f;
typedef __attribute__((ext_vector_type(8))) float v8f;

// Skeleton edge endpoints packed as 4-bit nibbles (edge i -> nibble i).
// src = [1,1,2,3,5,6,1,8,9,10,11,9,13,14,0]
// dst = [2,5,3,4,6,7,8,9,10,11,12,13,14,15,1]
#define SRC_PACK 0x00ED9BA981653211ULL
#define DST_PACK 0x01FEDCBA98764352ULL

__global__ __launch_bounds__(32) void
skel_loss_kernel(const float* __restrict__ y_pred3d,  // 16 x 3
                 const float* __restrict__ y_data,    // 16 x 2
                 const float* __restrict__ z,         // unused
                 const float* __restrict__ rleg,      // 2
                 const float* __restrict__ lleg,      // 2
                 float* __restrict__ out)             // 7
{
    const int lane = threadIdx.x;            // 0..31, one wave32
    const float EPS = 1e-12f;

    // ---- per-edge terms on lanes 0..14 ----
    float stan = 0.0f, spar = 0.0f;
    if (lane < 15) {
        const int s = (int)((SRC_PACK >> (4 * lane)) & 0xF);
        const int d = (int)((DST_PACK >> (4 * lane)) & 0xF);

        const float bx = y_data[2 * d + 0] - y_data[2 * s + 0];
        const float by = y_data[2 * d + 1] - y_data[2 * s + 1];
        const float nb = sqrtf(bx * bx + by * by);
        const float inv = 1.0f / fmaxf(nb, EPS);
        const float tx = bx * inv, ty = by * inv;   // tangent
        const float nx = -ty, ny = tx;              // normal

        float ax = y_pred3d[3 * d + 0] - y_pred3d[3 * s + 0];
        float ay = y_pred3d[3 * d + 1] - y_pred3d[3 * s + 1];
        if (lane == 9)  { ax = rleg[0]; ay = rleg[1]; }   // RLEG_IDX
        if (lane == 12) { ax = lleg[0]; ay = lleg[1]; }   // LLEG_IDX

        const float cx = ax - bx, cy = ay - by;
        const float dt = cx * tx + cy * ty;
        const float dn = cx * nx + cy * ny;
        stan = dt * dt;
        spar = dn * dn;
    }

    // ---- ground terms on lanes 0 and 1 ----
    float gterm = 0.0f;
    if (lane < 2) {
        const int g0 = (lane == 0) ? 11 : 14;
        const int g1 = (lane == 0) ? 12 : 15;
        const float gx = y_pred3d[3 * g1 + 0] - y_pred3d[3 * g0 + 0];
        const float gy = y_pred3d[3 * g1 + 1] - y_pred3d[3 * g0 + 1];
        const float gn = sqrtf(gx * gx + gy * gy);
        const float gxn = gx / fmaxf(gn, EPS);
        const float e = fabsf(gxn) - 1.0f;
        gterm = e * e;
    }
    const float gsum = gterm + __shfl_xor(gterm, 1, 32); // lane0 <- lane0+lane1

    // ---- WMMA f32 16x16x4 as a one-shot cross-lane reduction ----
    // A[m][k]: rows 0..3 hold tan terms (4 per row along K), rows 4..7 par
    // terms; B = all-ones -> D[m][n] = row-sum of A, broadcast to all n.
    // 32-bit A 16x4 layout: lane = m + 16*(k>=2), vgpr = k&1.
    const int m  = lane & 15;
    const int kh = lane >> 4;                 // 0 -> K{0,1}, 1 -> K{2,3}
    const int base = ((m & 3) << 2) + (kh << 1);

    const float t0 = __shfl(stan, base, 32);
    const float t1 = __shfl(stan, base + 1, 32);
    const float p0 = __shfl(spar, base, 32);
    const float p1 = __shfl(spar, base + 1, 32);

    v2f A;
    A.x = (m < 4) ? t0 : ((m < 8) ? p0 : 0.0f);
    A.y = (m < 4) ? t1 : ((m < 8) ? p1 : 0.0f);
    v2f B;
    B.x = 1.0f;
    B.y = 1.0f;
    v8f C = {};
    // All lanes active here: EXEC is all ones as WMMA requires.
    C = __builtin_amdgcn_wmma_f32_16x16x4_f32(
            /*neg_a=*/false, A, /*neg_b=*/false, B,
            /*c_mod=*/(short)0, C, /*reuse_a=*/false, /*reuse_b=*/false);

    // C/D layout: vgpr r, lanes 0-15 -> M=r. Lane 0 holds all 8 group sums.
    if (lane == 0) {
        const float Ltan = (C[0] + C[1] + C[2] + C[3]) * (1.0f / 15.0f);
        const float Lpar = (C[4] + C[5] + C[6] + C[7]) * (1.0f / 15.0f);
        const float Lgr  = gsum * 0.5f;
        out[0] = Ltan;
        out[1] = 0.0f;
        out[2] = Lpar;
        out[3] = 0.0f;
        out[4] = Lgr;
        out[5] = 0.0f;
        out[6] = 0.0f;
    }
    (void)z;
}

extern "C" void kernel_launch(void* const* d_in, const int* in_sizes, int n_in,
                              void* d_out, int out_size, void* d_ws, size_t ws_size,
                              hipStream_t stream) {
    (void)in_sizes; (void)n_in; (void)out_size; (void)d_ws; (void)ws_size;
    const float* y_pred3d = (const float*)d_in[0];
    const float* y_data   = (const float*)d_in[1];
    const float* z        = (const float*)d_in[2];
    const float* rleg     = (const float*)d_in[3];
    const float* lleg     = (const float*)d_in[4];
    skel_loss_kernel<<<1, 32, 0, stream>>>(y_pred3d, y_data, z, rleg, lleg,
                                           (float*)d_out);
}